// SoftDTW_12017318494223
// MI455X (gfx1250) — compile-verified
//
#include <hip/hip_runtime.h>

typedef __attribute__((ext_vector_type(2))) float v2f;
typedef __attribute__((ext_vector_type(8))) float v8f;

#define BATCH 64
#define NSEQ  512
#define DDIM  64
#define BIGF  1e30f
#define CH2   8    // diagonals per LDS-staged chunk in the DP kernel (double-buffered)

// ---- CDNA5 async global->LDS path (guarded; falls back to sync staging) ----
#if defined(__gfx1250__) && __has_builtin(__builtin_amdgcn_global_load_async_to_lds_b32)
#  define USE_ASYNC_LDS 1
#else
#  define USE_ASYNC_LDS 0
#endif

#if defined(__gfx1250__) && __has_builtin(__builtin_amdgcn_s_wait_asynccnt)
#  define ASYNC_WAIT() __builtin_amdgcn_s_wait_asynccnt(0)
#elif defined(__gfx1250__)
#  define ASYNC_WAIT() asm volatile("s_wait_asynccnt 0x0" ::: "memory")
#else
#  define ASYNC_WAIT()
#endif

// b32 async builtin takes (int AS1*, int AS3*, imm offset, imm cpol)
#define TO_GLOBAL(p) ((__attribute__((address_space(1))) int*)(p))
#define TO_LDS(p)    ((__attribute__((address_space(3))) int*)(p))

// ---------------------------------------------------------------------------
// Kernel 1: D[b] = xx + yy - 2 * X[b] @ Y[b]^T   via V_WMMA_F32_16X16X4_F32
// block = 256 threads (8 waves), grid = (batch=64, row-slab=8)
// Each workgroup computes a 64-row x 512-col slab of D (row-major, coalesced).
// ---------------------------------------------------------------------------
__global__ __launch_bounds__(256) void sqdist_wmma(const float* __restrict__ X,
                                                   const float* __restrict__ Y,
                                                   float* __restrict__ D) {
    __shared__ float Xs[64][68];   // 64 rows x 64 k (padded)
    __shared__ float Ys[64][68];   // 64 cols x 64 k (padded)
    __shared__ float xxs[64];
    __shared__ float yys[64];

    const int b   = blockIdx.x;
    const int R0  = blockIdx.y * 64;
    const int tid = threadIdx.x;

    const float* Xb = X + ((size_t)b * NSEQ + R0) * DDIM;

    // Stage X slab: coalesced (consecutive tid -> consecutive k)
    for (int e = tid; e < 64 * 64; e += 256) {
        int r = e >> 6, k = e & 63;
        Xs[r][k] = Xb[r * DDIM + k];
    }
    __syncthreads();
    if (tid < 64) {
        float s = 0.f;
        for (int k = 0; k < 64; ++k) { float v = Xs[tid][k]; s += v * v; }
        xxs[tid] = s;
    }

    const int wave  = tid >> 5;
    const int lane  = tid & 31;
    const int rt    = wave & 3;       // row tile 0..3 within slab
    const int cpair = wave >> 2;      // 0 -> col tiles {0,1}, 1 -> {2,3}
    const int lm    = lane & 15;
    const int lh    = lane >> 4;      // lane half (selects K pair / M+8)
    const int ct0   = cpair * 2;
    const int ct1   = cpair * 2 + 1;

    float* Dbase = D + (size_t)b * NSEQ * NSEQ;

    for (int cp = 0; cp < 8; ++cp) {      // 8 column panels of 64
        const int C0 = cp * 64;
        __syncthreads();                  // previous panel fully consumed
        const float* Yb = Y + ((size_t)b * NSEQ + C0) * DDIM;
        for (int e = tid; e < 64 * 64; e += 256) {
            int c = e >> 6, k = e & 63;
            Ys[c][k] = Yb[c * DDIM + k];
        }
        __syncthreads();
        if (tid < 64) {
            float s = 0.f;
            for (int k = 0; k < 64; ++k) { float v = Ys[tid][k]; s += v * v; }
            yys[tid] = s;
        }
        __syncthreads();

        v8f acc0 = {};
        v8f acc1 = {};
        // K loop: 16 steps of K=4 (f32 WMMA)
        for (int k0 = 0; k0 < DDIM; k0 += 4) {
            const int kb = k0 + (lh << 1);    // lanes 16-31 hold K = k0+2,k0+3
            v2f a, b0, b1;
            a.x  = Xs[rt * 16 + lm][kb];
            a.y  = Xs[rt * 16 + lm][kb + 1];
            b0.x = Ys[ct0 * 16 + lm][kb];
            b0.y = Ys[ct0 * 16 + lm][kb + 1];
            b1.x = Ys[ct1 * 16 + lm][kb];
            b1.y = Ys[ct1 * 16 + lm][kb + 1];
            acc0 = __builtin_amdgcn_wmma_f32_16x16x4_f32(false, a, false, b0,
                                                         (short)0, acc0, false, false);
            acc1 = __builtin_amdgcn_wmma_f32_16x16x4_f32(false, a, false, b1,
                                                         (short)0, acc1, false, false);
        }

        // Fused epilogue: D = xx + yy - 2*S, coalesced row-major stores.
        const int gc0 = C0 + ct0 * 16 + lm;
        const int gc1 = C0 + ct1 * 16 + lm;
        const float yv0 = yys[ct0 * 16 + lm];
        const float yv1 = yys[ct1 * 16 + lm];
        for (int v = 0; v < 8; ++v) {
            const int lr = rt * 16 + v + (lh << 3);   // C layout: lanes>=16 hold M+8
            const int gr = R0 + lr;
            const float xv = xxs[lr];
            Dbase[(size_t)gr * NSEQ + gc0] = xv + yv0 - 2.0f * acc0[v];
            Dbase[(size_t)gr * NSEQ + gc1] = xv + yv1 - 2.0f * acc1[v];
        }
    }
}

// ---------------------------------------------------------------------------
// Kernel 2: soft-DTW anti-diagonal wavefront. One workgroup per batch,
// 512 threads; thread t owns row i = t+1. Diagonal buffers rotate in LDS.
// D is consumed in 8-diagonal bands, double-buffered in LDS: the next band
// is fetched with GLOBAL_LOAD_ASYNC_TO_LDS (ASYNCcnt) while the current
// band's DP steps run, then s_wait_asynccnt 0 + barrier before the swap.
// ---------------------------------------------------------------------------
__global__ __launch_bounds__(512) void softdtw_dp(const float* __restrict__ D,
                                                  float* __restrict__ out) {
    __shared__ float rr[3][520];               // 3 rotating wavefront buffers
    __shared__ float Dbuf[2][512][CH2 + 1];    // odd stride 9 -> conflict-free

    const int b = blockIdx.x;
    const int t = threadIdx.x;                 // row r = t, cell i = t+1
    const float* Db = D + (size_t)b * NSEQ * NSEQ;

    rr[0][t + 1] = BIGF;                       // r0
    rr[1][t + 1] = BIGF;                       // r1
    if (t == 0) { rr[0][0] = 0.0f; rr[1][0] = BIGF; rr[2][0] = BIGF; }

    // Band loader: Dbuf[sel][row][k] = D[row][d0 + k - row - 2]
    auto stage = [&](int d0, int sel) {
        for (int e = t; e < NSEQ * CH2; e += NSEQ) {
            const int row = e >> 3;            // CH2 == 8
            const int k   = e & (CH2 - 1);
            const int c   = d0 + k - row - 2;
            if ((unsigned)c < (unsigned)NSEQ) {
#if USE_ASYNC_LDS
                __builtin_amdgcn_global_load_async_to_lds_b32(
                    TO_GLOBAL(Db + row * NSEQ + c),
                    TO_LDS(&Dbuf[sel][row][k]), 0, 0);
#else
                Dbuf[sel][row][k] = Db[row * NSEQ + c];
#endif
            }
        }
    };

    int sel = 0;
    stage(2, 0);                               // prefetch first band
    ASYNC_WAIT();
    __syncthreads();

    for (int d0 = 2; d0 <= 2 * NSEQ; d0 += CH2) {
        const int dend = min(d0 + CH2, 2 * NSEQ + 1);
        const int dnext = d0 + CH2;
        if (dnext <= 2 * NSEQ) stage(dnext, sel ^ 1);   // async prefetch next band

        for (int d = d0; d < dend; ++d) {
            const int i2 = (d - 2) % 3;        // diagonal d-2
            const int i1 = (d - 1) % 3;        // diagonal d-1
            const int in = d % 3;              // new diagonal

            const float a  = rr[i2][t];        // R[i-1][j-1]
            const float bb = rr[i1][t];        // R[i-1][j]
            const float cc = rr[i1][t + 1];    // R[i][j-1]

            const float zmax = fmaxf(fmaxf(-a, -bb), -cc);
            const float sum  = __expf(-a - zmax) + __expf(-bb - zmax) + __expf(-cc - zmax);
            const float softmin = -(__logf(sum) + zmax);

            const int c = d - t - 2;           // j-1
            float rnew = BIGF;
            if ((unsigned)c < (unsigned)NSEQ)
                rnew = Dbuf[sel][t][d - d0] + softmin;

            rr[in][t + 1] = rnew;
            if (t == 0) rr[in][0] = BIGF;
            if (d == 2 * NSEQ && t == NSEQ - 1) out[b] = rnew;   // r_last[:, N]
            __syncthreads();
        }

        ASYNC_WAIT();                          // next band landed in LDS
        __syncthreads();
        sel ^= 1;
    }
}

// ---------------------------------------------------------------------------
extern "C" void kernel_launch(void* const* d_in, const int* in_sizes, int n_in,
                              void* d_out, int out_size, void* d_ws, size_t ws_size,
                              hipStream_t stream) {
    const float* X = (const float*)d_in[0];
    const float* Y = (const float*)d_in[1];
    float* out = (float*)d_out;
    float* Dws = (float*)d_ws;   // needs 64*512*512*4 = 64 MiB scratch

    sqdist_wmma<<<dim3(BATCH, NSEQ / 64), 256, 0, stream>>>(X, Y, Dws);
    softdtw_dp<<<BATCH, NSEQ, 0, stream>>>(Dws, out);
}